// Eikonal2D_7189775253698
// MI455X (gfx1250) — compile-verified
//
#include <hip/hip_runtime.h>
#include <math.h>

// ---------------------------------------------------------------------------
// Eikonal2D on MI455X (gfx1250, wave32)
//
// Kernel 1 (eik_solve): one 1024-thread workgroup owns the whole 256x256 grid
//   in LDS (258x258 halo = 266 KB; CDNA5's 320 KB/WGP makes this possible).
//   Slowness f is staged global->LDS with GLOBAL_LOAD_ASYNC_TO_LDS_B32
//   (ASYNCcnt path), pulled into 64 registers/thread, then the LDS buffer is
//   reused as the travel-time table u. 512 Godunov sweeps, one barrier each.
//
// Kernel 2 (interp_picks): each block stages the converged 256 KB table into
//   LDS with a SINGLE Tensor-Data-Mover DMA (tensor_load_to_lds + TENSORcnt),
//   then does bandwidth-bound bilinear gathers for its slice of the 2M picks.
//   Pick streams are B128 vectors + non-temporal (read/written exactly once;
//   keeps L2 free for the table). No WMMA: the 4-wide bilinear dot product
//   has no matrix structure worth a 16x16 tile (15/16 waste > 7 FMAs).
// ---------------------------------------------------------------------------

#define NXg 256
#define NYg 256
#define NCELL (NXg * NYg)          // 65536 cells, 256 KB as f32
#define N_ITERS 512
#define BIGF 1.0e6f

#define HSTRIDE 258                // halo row stride
#define HSIZE (258 * 258)          // 66564 words = 266,256 B  (< 320 KB LDS)

#define PICKS_PER_BLOCK 16384
#define GROUPS_PER_BLOCK (PICKS_PER_BLOCK / 4)

typedef unsigned int v4u __attribute__((ext_vector_type(4)));
typedef int          v4i __attribute__((ext_vector_type(4)));
typedef int          v8i __attribute__((ext_vector_type(8)));
typedef float        v4f __attribute__((ext_vector_type(4)));  // native clang
                                                               // vector: legal
                                                               // for NT builtins

// --- CDNA5 async global->LDS copy (ASYNCcnt), per CDNA5 ISA ---
__device__ __forceinline__ void async_g2lds_b32(unsigned lds_byte_off,
                                                unsigned g_byte_off,
                                                const void* base) {
  asm volatile("global_load_async_to_lds_b32 %0, %1, %2"
               :
               : "v"(lds_byte_off), "v"(g_byte_off), "s"(base)
               : "memory");
}

__device__ __forceinline__ void wait_async0() {
  asm volatile("s_wait_asynccnt 0" ::: "memory");
}

// Low 32 bits of a generic pointer to __shared__ == LDS byte offset
// (aperture layout: addr[31:0] is the LDS offset; HW adds LDS_BASE).
__device__ __forceinline__ unsigned lds_off(const void* p) {
  return (unsigned)(unsigned long long)p;
}

// --- Tensor Data Mover: one-shot 2D tile (256x256 f32) global -> LDS -------
// D# per CDNA5 ISA (groups 2/3 = 0 for a
// 2D tensor). Issued by a single wave; TDM ignores EXEC.
__device__ __forceinline__ void tdm_load_table_to_lds(const void* gbase,
                                                      unsigned lds_byte) {
  const unsigned long long ga = (unsigned long long)gbase;
  v4u g0;
  g0.x = 1u;                                   // count=1, no gather, user desc
  g0.y = lds_byte;                             // lds_addr [63:32]
  g0.z = (unsigned)ga;                         // global_addr [95:64]
  g0.w = ((unsigned)(ga >> 32) & 0x01FFFFFFu)  // global_addr [120:96]
         | 0x80000000u;                        // type=2 ("image") [127:126]
  v8i g1;
  g1[0] = 0x00020000;      // workgroup_mask=0, data_size=2 (4 bytes)
  g1[1] = 0x01000000;      // tensor_dim0=256  (bits [79:48])
  g1[2] = 0x01000000;      // tensor_dim1=256  (bits [111:80])
  g1[3] = 0x01000000;      // tile_dim0=256    (bits [127:112])
  g1[4] = 0x00000100;      // tile_dim1=256, tile_dim2=0
  g1[5] = 256;             // tensor_dim0_stride low32 = 256 elements
  g1[6] = 0;               // stride0 hi16 = 0, stride1 low16 = 0
  g1[7] = 1;               // tensor_dim1_stride = 65536 (bits [255:224] = 1)
  v4i gz = {0, 0, 0, 0};
#if __clang_major__ >= 23
  v8i gz8 = {0, 0, 0, 0, 0, 0, 0, 0};
  __builtin_amdgcn_tensor_load_to_lds(g0, g1, gz, gz, gz8, 0);
#else
  __builtin_amdgcn_tensor_load_to_lds(g0, g1, gz, gz, 0);
#endif
}

// ---------------------------------------------------------------------------
__global__ __launch_bounds__(1024) void eik_solve(
    const float* __restrict__ f, const float* __restrict__ src_xy,
    float* __restrict__ u_out) {
  __shared__ float smem[HSIZE];

  const int tid = threadIdx.x;
  const int j   = tid & 255;   // column (stride-1 across lanes: conflict-free)
  const int r0  = tid >> 8;    // row phase 0..3; thread owns rows r0+4k, k<64

  // ---- stage f (256 KB) into LDS front portion via async bulk copy --------
#pragma unroll 4
  for (int k = 0; k < 64; ++k) {
    const int w = tid + (k << 10);
    async_g2lds_b32(lds_off(&smem[w]), (unsigned)(w << 2), f);
  }
  wait_async0();
  __syncthreads();

  // ---- per-cell slowness*H into registers (H == 1.0) ----------------------
  float fh[64];
#pragma unroll
  for (int k = 0; k < 64; ++k) fh[k] = smem[tid + (k << 10)];

  // ---- source seeding (4 cells around fractional source) ------------------
  const float sx = src_xy[0], sy = src_xy[1];
  const int ix0 = (int)fminf(fmaxf(floorf(sx), 0.f), (float)(NXg - 2));
  const int iy0 = (int)fminf(fmaxf(floorf(sy), 0.f), (float)(NYg - 2));
  float seedval = 0.f;
  int   seedidx = 0;
  if (tid < 4) {
    const int ii = ix0 + (tid >> 1);
    const int jj = iy0 + (tid & 1);
    const float dx = (float)ii - sx;
    const float dy = (float)jj - sy;
    seedval = sqrtf(dx * dx + dy * dy) * smem[ii * NYg + jj];  // * H * f[ii,jj]
    seedidx = (ii + 1) * HSTRIDE + (jj + 1);
  }
  __syncthreads();  // all reads of f done before reusing LDS as u

  // ---- init u = BIG everywhere (halo included), place seeds ---------------
  for (int w = tid; w < HSIZE; w += 1024) smem[w] = BIGF;
  __syncthreads();
  if (tid < 4) smem[seedidx] = seedval;
  __syncthreads();

  // ---- 512 Godunov upwind sweeps (min-monotone fixed point) ---------------
#pragma unroll 1
  for (int it = 0; it < N_ITERS; ++it) {
#pragma unroll
    for (int k = 0; k < 64; ++k) {
      const int r    = r0 + (k << 2);
      const int base = (r + 1) * HSTRIDE + (j + 1);
      const float up = smem[base - HSTRIDE];
      const float dn = smem[base + HSTRIDE];
      const float lt = smem[base - 1];
      const float rt = smem[base + 1];
      const float uo = smem[base];
      const float a  = fminf(up, dn);
      const float b  = fminf(lt, rt);
      const float fhk = fh[k];
      const float one_sided = fminf(a, b) + fhk;
      const float d  = a - b;
      const float dd = d * d;
      const float disc = fmaxf(fmaf(fhk + fhk, fhk, -dd), 0.f);
      const float two_sided = 0.5f * (a + b + sqrtf(disc));
      const float cand = (fabsf(d) >= fhk) ? one_sided : two_sided;
      smem[base] = fminf(uo, cand);
    }
    __syncthreads();
  }

  // ---- write converged travel-time table to global scratch ----------------
#pragma unroll
  for (int k = 0; k < 64; ++k) {
    const int r = r0 + (k << 2);
    u_out[(r << 8) + j] = smem[(r + 1) * HSTRIDE + (j + 1)];
  }
}

// ---------------------------------------------------------------------------
// Bilinear sample of the LDS-resident table, matching the reference's
// floor+clamp semantics exactly.
__device__ __forceinline__ float bilerp_lds(const float* tt, float x, float y) {
  const float fx = fminf(fmaxf(floorf(x), 0.f), (float)(NXg - 2));
  const float fy = fminf(fmaxf(floorf(y), 0.f), (float)(NYg - 2));
  const int ixl = (int)fx;
  const int iyl = (int)fy;
  const float xc = fminf(fmaxf(x, 0.f), (float)(NXg - 1));
  const float yc = fminf(fmaxf(y, 0.f), (float)(NYg - 1));
  const float wx0 = xc - fx;          // (xc - ix0)
  const float wx1 = 1.f - wx0;        // (ix1 - xc)
  const float wy0 = yc - fy;
  const float wy1 = 1.f - wy0;
  const int b = (ixl << 8) + iyl;
  const float Q00 = tt[b];
  const float Q01 = tt[b + 1];
  const float Q10 = tt[b + NYg];
  const float Q11 = tt[b + NYg + 1];
  return Q00 * wx1 * wy1 + Q10 * wx0 * wy1 +
         Q01 * wx1 * wy0 + Q11 * wx0 * wy0;
}

__global__ __launch_bounds__(1024) void interp_picks(
    const float* __restrict__ tt_g, const float* __restrict__ ex,
    const float* __restrict__ ey, const float* __restrict__ et,
    float* __restrict__ out, int n) {
  __shared__ float tt[NCELL];  // 256 KB table, LDS-resident per block

  const int tid = threadIdx.x;

  // One TDM DMA moves the whole table; only wave 0 reaches the issue point
  // (other waves branch around on execz), TDM itself ignores EXEC.
  if (tid == 0) {
    tdm_load_table_to_lds(tt_g, lds_off(&tt[0]));
  }
  __builtin_amdgcn_s_wait_tensorcnt(0);   // wave 0 gates on DMA completion
  __syncthreads();                        // ...then releases the block

  // ---- B128 non-temporal streaming over full pick quads -------------------
  const int ngroups = n >> 2;
  const int gstart  = blockIdx.x * GROUPS_PER_BLOCK;
  int gend = gstart + GROUPS_PER_BLOCK;
  if (gend > ngroups) gend = ngroups;

  const v4f* ex4 = (const v4f*)ex;
  const v4f* ey4 = (const v4f*)ey;
  const v4f* et4 = (const v4f*)et;
  v4f* out4 = (v4f*)out;

  for (int g = gstart + tid; g < gend; g += 1024) {
    const v4f x4 = __builtin_nontemporal_load(&ex4[g]);
    const v4f y4 = __builtin_nontemporal_load(&ey4[g]);
    const v4f t4 = __builtin_nontemporal_load(&et4[g]);
    v4f r;
    r.x = t4.x + bilerp_lds(tt, x4.x, y4.x);
    r.y = t4.y + bilerp_lds(tt, x4.y, y4.y);
    r.z = t4.z + bilerp_lds(tt, x4.z, y4.z);
    r.w = t4.w + bilerp_lds(tt, x4.w, y4.w);
    __builtin_nontemporal_store(r, &out4[g]);
  }

  // ---- scalar tail (n % 4 picks), handled once by block 0 -----------------
  if (blockIdx.x == 0) {
    for (int p = (ngroups << 2) + tid; p < n; p += 1024) {
      const float x = __builtin_nontemporal_load(&ex[p]);
      const float y = __builtin_nontemporal_load(&ey[p]);
      const float t = __builtin_nontemporal_load(&et[p]);
      __builtin_nontemporal_store(t + bilerp_lds(tt, x, y), &out[p]);
    }
  }
}

// ---------------------------------------------------------------------------
extern "C" void kernel_launch(void* const* d_in, const int* in_sizes, int n_in,
                              void* d_out, int out_size, void* d_ws,
                              size_t ws_size, hipStream_t stream) {
  const float* f      = (const float*)d_in[0];  // [256*256] slowness
  const float* src_xy = (const float*)d_in[1];  // [2]
  const float* ex     = (const float*)d_in[2];  // [N]
  const float* ey     = (const float*)d_in[3];  // [N]
  const float* et     = (const float*)d_in[4];  // [N]
  float* out = (float*)d_out;
  float* u   = (float*)d_ws;                    // 256 KB travel-time table
  const int n = in_sizes[2];

  eik_solve<<<1, 1024, 0, stream>>>(f, src_xy, u);

  const int nblk = (n + PICKS_PER_BLOCK - 1) / PICKS_PER_BLOCK;
  interp_picks<<<nblk, 1024, 0, stream>>>(u, ex, ey, et, out, n);
}